// GCNModel_62036507623981
// MI455X (gfx1250) — compile-verified
//
#include <hip/hip_runtime.h>

typedef __attribute__((ext_vector_type(2))) float v2f;
typedef __attribute__((ext_vector_type(8))) float v8f;

#define FDIM 128

// ---------------------------------------------------------------------------
// Degree / normalization kernels
// ---------------------------------------------------------------------------
__global__ void init_deg_kernel(float* __restrict__ deg, int n) {
  int i = blockIdx.x * blockDim.x + threadIdx.x;
  if (i < n) deg[i] = 1.0f;  // self-loop contributes 1 to every node's degree
}

__global__ void accum_deg_kernel(const long long* __restrict__ dst,
                                 float* __restrict__ deg, long long E) {
  long long e = (long long)blockIdx.x * blockDim.x + threadIdx.x;
  if (e < E) atomicAdd(&deg[(int)dst[e]], 1.0f);
}

__global__ void rsqrt_kernel(float* __restrict__ deg, int n) {
  int i = blockIdx.x * blockDim.x + threadIdx.x;
  if (i < n) {
    float d = deg[i];
    deg[i] = (d > 0.0f) ? rsqrtf(d) : 0.0f;  // deg >= 1 here, guard for parity
  }
}

// ---------------------------------------------------------------------------
// Seed the aggregation buffer with the self-loop term + bias:
//   out[i, f] = h[i, f] * dinv[i]^2 + bias[f]
// (replaces a plain zero-fill; the edge scatter then atomically accumulates
//  on top of this, so no separate finalize pass is needed.)
// ---------------------------------------------------------------------------
__global__ void seed_kernel(float4* __restrict__ out, const float4* __restrict__ h,
                            const float* __restrict__ dinv,
                            const float4* __restrict__ bias, long long n4) {
  long long i = (long long)blockIdx.x * blockDim.x + threadIdx.x;
  if (i >= n4) return;
  const int node = (int)(i >> 5);            // 32 float4 per 128-wide row
  const int f4 = (int)(i & 31);
  const float di = dinv[node];
  const float s = di * di;
  const float4 hv = h[i];
  const float4 b = bias[f4];
  out[i] = make_float4(fmaf(hv.x, s, b.x), fmaf(hv.y, s, b.y),
                       fmaf(hv.z, s, b.z), fmaf(hv.w, s, b.w));
}

// ---------------------------------------------------------------------------
// Dense transform  H = relu?(X) @ W  via full-precision fp32 WMMA (16x16x4).
//   block = 256 threads = 8 waves; blockIdx.x selects a 16-row stripe;
//   wave w computes output columns [16w, 16w+16).
// RELU is fused into the A-fragment loads: the GCN activation relu(agg) is
// only ever consumed here, so no separate activation pass is required.
// A frag (16x4 f32, 2 VGPRs): lanes 0-15 -> M=lane, K={k,k+1};
//                             lanes 16-31 -> M=lane-16, K={k+2,k+3}.
// B frag (4x16 f32, 2 VGPRs): mirrored row-per-half-wave layout.
// C/D (16x16 f32, 8 VGPRs):   VGPR j -> M = j + 8*(lane>=16), N = lane&15.
// ---------------------------------------------------------------------------
template <bool RELU>
__global__ __launch_bounds__(256)
void gemm_wmma_f32(const float* __restrict__ X, const float* __restrict__ W,
                   float* __restrict__ H, int nRows) {
  const int wave = threadIdx.x >> 5;
  const int lane = threadIdx.x & 31;
  const int half = lane >> 4;       // 0: lanes 0-15, 1: lanes 16-31
  const int m    = lane & 15;
  const int rowBase = blockIdx.x << 4;
  const int col = (wave << 4) + m;  // output column (also B/C N index)

  int rload = rowBase + m;
  if (rload >= nRows) rload = nRows - 1;     // clamp tail reads
  const float* xrow = X + (size_t)rload * FDIM;

  v8f c = {0.f, 0.f, 0.f, 0.f, 0.f, 0.f, 0.f, 0.f};

#pragma unroll
  for (int k = 0; k < FDIM; k += 4) {
    const int ka = k + (half << 1);
    v2f a;
    a.x = xrow[ka];
    a.y = xrow[ka + 1];
    if (RELU) {
      a.x = fmaxf(a.x, 0.0f);
      a.y = fmaxf(a.y, 0.0f);
    }
    v2f b;
    b.x = W[(size_t)ka * FDIM + col];
    b.y = W[(size_t)(ka + 1) * FDIM + col];
    c = __builtin_amdgcn_wmma_f32_16x16x4_f32(
        /*neg_a=*/false, a, /*neg_b=*/false, b,
        /*c_mod=*/(short)0, c, /*reuse_a=*/false, /*reuse_b=*/false);
  }

#pragma unroll
  for (int j = 0; j < 8; ++j) {
    const int mr = rowBase + (half << 3) + j;
    if (mr < nRows) H[(size_t)mr * FDIM + col] = c[j];
  }
}

// ---------------------------------------------------------------------------
// Edge scatter: one wave per edge. Gather the 512B source row (float4/lane),
// scale by norm = dinv[src]*dinv[dst], fp32 atomic-add into the dest row.
// Feature tables are 51.2MB -> L2 resident on MI455X (192MB L2).
// ---------------------------------------------------------------------------
__global__ __launch_bounds__(256)
void scatter_kernel(const long long* __restrict__ src,
                    const long long* __restrict__ dst,
                    const float* __restrict__ dinv,
                    const float* __restrict__ h,
                    float* __restrict__ out, long long E) {
  long long e = (long long)blockIdx.x * 8 + (threadIdx.x >> 5);
  if (e >= E) return;
  const int lane = threadIdx.x & 31;
  const int s = (int)src[e];
  const int d = (int)dst[e];
  const float nrm = dinv[s] * dinv[d];
  const float4 v = ((const float4*)(h + (size_t)s * FDIM))[lane];
  float* orow = out + (size_t)d * FDIM + (lane << 2);
  atomicAdd(orow + 0, v.x * nrm);
  atomicAdd(orow + 1, v.y * nrm);
  atomicAdd(orow + 2, v.z * nrm);
  atomicAdd(orow + 3, v.w * nrm);
}

// ---------------------------------------------------------------------------
// Host-side launch sequence (graph-capture safe; stream-ordered only).
// Inputs: x, edge_index(int64, [2,E]), W1, b1, W2, b2, W3, b3.
// Per layer:  h = relu?(prev) @ W   (WMMA fp32)
//             out = h*dinv^2 + b    (seed, self-loop term)
//             out[dst] += h[src]*dinv[s]*dinv[d]  (edge scatter, atomics)
// ReLU of layers 1/2 is folded into the next layer's GEMM A-loads; layer 3
// has no ReLU, so its scatter output is final.
// ---------------------------------------------------------------------------
extern "C" void kernel_launch(void* const* d_in, const int* in_sizes, int n_in,
                              void* d_out, int out_size, void* d_ws, size_t ws_size,
                              hipStream_t stream) {
  const float*     x   = (const float*)d_in[0];
  const long long* ei  = (const long long*)d_in[1];
  const float*     W1  = (const float*)d_in[2];
  const float*     b1  = (const float*)d_in[3];
  const float*     W2  = (const float*)d_in[4];
  const float*     b2  = (const float*)d_in[5];
  const float*     W3  = (const float*)d_in[6];
  const float*     b3  = (const float*)d_in[7];
  float* out = (float*)d_out;

  const int       N = in_sizes[0] / FDIM;
  const long long E = (long long)in_sizes[1] / 2;
  const long long* srcIdx = ei;
  const long long* dstIdx = ei + E;

  // Workspace: [deg/dinv: N][h: N*128][a: N*128]  (~103 MB)
  float* deg = (float*)d_ws;
  float* h   = deg + (((size_t)N + 63) & ~(size_t)63);
  float* a   = h + (size_t)N * FDIM;

  const long long total = (long long)N * FDIM;
  const long long n4 = total / 4;
  const int degGrid  = (N + 255) / 256;
  const int edgGrid  = (int)((E + 255) / 256);
  const int gemmGrid = (N + 15) / 16;
  const int seedGrid = (int)((n4 + 255) / 256);
  const int scatGrid = (int)((E + 7) / 8);

  // Symmetric normalization coefficients.
  init_deg_kernel<<<degGrid, 256, 0, stream>>>(deg, N);
  accum_deg_kernel<<<edgGrid, 256, 0, stream>>>(dstIdx, deg, E);
  rsqrt_kernel<<<degGrid, 256, 0, stream>>>(deg, N);   // deg -> dinv in place

  // Layer 1: h = x@W1; a = h*dinv^2 + b1 + scatter
  gemm_wmma_f32<false><<<gemmGrid, 256, 0, stream>>>(x, W1, h, N);
  seed_kernel<<<seedGrid, 256, 0, stream>>>((float4*)a, (const float4*)h, deg,
                                            (const float4*)b1, n4);
  scatter_kernel<<<scatGrid, 256, 0, stream>>>(srcIdx, dstIdx, deg, h, a, E);

  // Layer 2: h = relu(a)@W2; a = h*dinv^2 + b2 + scatter
  gemm_wmma_f32<true><<<gemmGrid, 256, 0, stream>>>(a, W2, h, N);
  seed_kernel<<<seedGrid, 256, 0, stream>>>((float4*)a, (const float4*)h, deg,
                                            (const float4*)b2, n4);
  scatter_kernel<<<scatGrid, 256, 0, stream>>>(srcIdx, dstIdx, deg, h, a, E);

  // Layer 3: h = relu(a)@W3; out = h*dinv^2 + b3 + scatter (no ReLU)
  gemm_wmma_f32<true><<<gemmGrid, 256, 0, stream>>>(a, W3, h, N);
  seed_kernel<<<seedGrid, 256, 0, stream>>>((float4*)out, (const float4*)h, deg,
                                            (const float4*)b3, n4);
  scatter_kernel<<<scatGrid, 256, 0, stream>>>(srcIdx, dstIdx, deg, h, out, E);
}